// Decoder_867583394600
// MI455X (gfx1250) — compile-verified
//
#include <hip/hip_runtime.h>
#include <hip/hip_bf16.h>

// ---------------- problem dims ----------------
#define HD   512
#define ED   256
#define VD   50000
#define BD   128
#define SD   200
#define K2H  1024          // 2H
#define K3H  1536          // 3H
#define KIN  1280          // 2H+E (GRU input)
#define KOUT 1792          // 3H+E (output proj K)

typedef __attribute__((ext_vector_type(2))) float v2f;
typedef __attribute__((ext_vector_type(8))) float v8f;

// ---------------- ws layout (floats) ----------------
#define WS_U2     0
#define WS_CTX    1024
#define WS_TOGRU  (WS_CTX + BD * K2H)            // 132096
#define WS_TOOUT  (WS_TOGRU + BD * KIN)          // 295936
// total = 525312 floats (~2.1 MB)

// ============================================================
// K1: u2[k] = sum_h v[h] * attn_w[h, H + k],  k in [0, 2H)
// (attn_b and the h0 part of the scores are softmax-invariant)
// ============================================================
__global__ __launch_bounds__(256) void k_u2(const float* __restrict__ attn_w,
                                            const float* __restrict__ vvec,
                                            float* __restrict__ u2) {
    int k = blockIdx.x * 256 + threadIdx.x;      // 0..1023
    float s = 0.f;
    #pragma unroll 8
    for (int h = 0; h < HD; ++h)
        s += vvec[h] * attn_w[h * K3H + HD + k];
    u2[k] = s;
}

// ============================================================
// K2: per batch b: scores = enc_out[b] @ u2 ; softmax over S ;
//     ctx[b] = attn @ enc_out[b]
// ============================================================
__global__ __launch_bounds__(256) void k_attn(const float* __restrict__ enc,
                                              const float* __restrict__ u2g,
                                              float* __restrict__ ctx) {
    __shared__ float u2[K2H];
    __shared__ float sc[SD + 24];
    const int b = blockIdx.x, tid = threadIdx.x;
    const int wave = tid >> 5, lane = tid & 31;
    const float* encb = enc + (size_t)b * SD * K2H;

    for (int i = tid; i < K2H; i += 256) u2[i] = u2g[i];
    __syncthreads();

    // ---- scores: one wave per row, shuffle-reduced dot of length 1024 ----
    for (int s = wave; s < SD; s += 8) {
        const float4* row = (const float4*)(encb + s * K2H);
        float p = 0.f;
        #pragma unroll
        for (int it = 0; it < 8; ++it) {
            int k4 = lane + it * 32;             // float4 index 0..255
            float4 e = row[k4];
            p += e.x * u2[4 * k4 + 0] + e.y * u2[4 * k4 + 1]
               + e.z * u2[4 * k4 + 2] + e.w * u2[4 * k4 + 3];
        }
        #pragma unroll
        for (int o = 16; o; o >>= 1) p += __shfl_down(p, o);
        if (lane == 0) sc[s] = p;
    }
    __syncthreads();

    // ---- softmax over 200 scores, done by wave 0 ----
    if (wave == 0) {
        float vals[7];
        float m = -3.0e38f;
        #pragma unroll
        for (int i = 0; i < 7; ++i) {
            int s = lane + i * 32;
            vals[i] = (s < SD) ? sc[s] : -3.0e38f;
            m = fmaxf(m, vals[i]);
        }
        #pragma unroll
        for (int o = 16; o; o >>= 1) m = fmaxf(m, __shfl_xor(m, o));
        float sum = 0.f;
        #pragma unroll
        for (int i = 0; i < 7; ++i) {
            int s = lane + i * 32;
            float e = (s < SD) ? __expf(vals[i] - m) : 0.f;
            vals[i] = e;
            sum += e;
        }
        #pragma unroll
        for (int o = 16; o; o >>= 1) sum += __shfl_xor(sum, o);
        float inv = 1.f / sum;
        #pragma unroll
        for (int i = 0; i < 7; ++i) {
            int s = lane + i * 32;
            if (s < SD) sc[s] = vals[i] * inv;
        }
    }
    __syncthreads();

    // ---- context: second pass over enc_out (L2-resident), coalesced in d ----
    float c0 = 0.f, c1 = 0.f, c2 = 0.f, c3 = 0.f;
    for (int s = 0; s < SD; ++s) {
        const float* r = encb + s * K2H + tid;
        float a = sc[s];
        c0 += a * r[0];
        c1 += a * r[256];
        c2 += a * r[512];
        c3 += a * r[768];
    }
    float* cb = ctx + (size_t)b * K2H;
    cb[tid]       = c0;
    cb[tid + 256] = c1;
    cb[tid + 512] = c2;
    cb[tid + 768] = c3;
}

// ============================================================
// K3: gather emb row; build to_gru = relu([emb_x, ctx]) and the
//     raw ctx/emb slices of to_out = [h_new | ctx | emb_x]
// ============================================================
__global__ __launch_bounds__(256) void k_prep(const int* __restrict__ x,
                                              const float* __restrict__ emb,
                                              const float* __restrict__ ctx,
                                              float* __restrict__ to_gru,
                                              float* __restrict__ to_out) {
    const int b = blockIdx.x, tid = threadIdx.x;
    const int xi = x[b];
    const float* er = emb + (size_t)xi * ED;
    float* tg = to_gru + (size_t)b * KIN;
    float* to = to_out + (size_t)b * KOUT;
    {   // emb slice (256 elements, one per thread)
        float e = er[tid];
        tg[tid] = fmaxf(e, 0.f);
        to[K3H + tid] = e;
    }
    for (int d = tid; d < K2H; d += 256) {
        float c = ctx[(size_t)b * K2H + d];
        tg[ED + d] = fmaxf(c, 0.f);
        to[HD + d] = c;
    }
}

// ============================================================
// K4: GRU cell. One block per b, thread j owns gates r/z/n of unit j.
// ============================================================
__global__ __launch_bounds__(512) void k_gru(const float* __restrict__ to_gru,
                                             const float* __restrict__ hidden,
                                             const float* __restrict__ w_ih,
                                             const float* __restrict__ w_hh,
                                             const float* __restrict__ b_ih,
                                             const float* __restrict__ b_hh,
                                             float* __restrict__ to_out,
                                             float* __restrict__ hnew_out) {
    __shared__ float xg[KIN];
    __shared__ float h0[HD];
    const int b = blockIdx.x, tid = threadIdx.x;
    for (int i = tid; i < KIN; i += 512) xg[i] = to_gru[(size_t)b * KIN + i];
    h0[tid] = hidden[(size_t)b * HD + tid];
    __syncthreads();

    float gi[3], gh[3];
    #pragma unroll
    for (int g = 0; g < 3; ++g) {
        const int j = tid + g * HD;
        const float4* wr = (const float4*)(w_ih + (size_t)j * KIN);
        float s = 0.f;
        #pragma unroll 4
        for (int k = 0; k < KIN / 4; ++k) {
            float4 w4 = wr[k];
            s += w4.x * xg[4 * k + 0] + w4.y * xg[4 * k + 1]
               + w4.z * xg[4 * k + 2] + w4.w * xg[4 * k + 3];
        }
        gi[g] = s + b_ih[j];
        const float4* hr = (const float4*)(w_hh + (size_t)j * HD);
        float t = 0.f;
        #pragma unroll 4
        for (int k = 0; k < HD / 4; ++k) {
            float4 w4 = hr[k];
            t += w4.x * h0[4 * k + 0] + w4.y * h0[4 * k + 1]
               + w4.z * h0[4 * k + 2] + w4.w * h0[4 * k + 3];
        }
        gh[g] = t + b_hh[j];
    }
    float r = 1.f / (1.f + __expf(-(gi[0] + gh[0])));
    float z = 1.f / (1.f + __expf(-(gi[1] + gh[1])));
    float n = tanhf(gi[2] + r * gh[2]);
    float hn = (1.f - z) * n + z * h0[tid];
    to_out[(size_t)b * KOUT + tid] = hn;
    hnew_out[(size_t)b * HD + tid] = hn;
}

// ============================================================
// K5: logits = to_out @ out_w^T + out_b  via V_WMMA_F32_16X16X4_F32
//   Block tile: 256 vocab rows x 128 batch cols. 8 waves; each wave
//   owns two 16-row vocab sub-tiles (B-fragment reuse x2) and 8 batch
//   tiles. B fragments come from LDS in a K-quad interleaved layout:
//   lds[(k/4)*512 + b*4 + (k%4)]  ->  one ds_load_b64 per fragment,
//   conflict-free on both read (banks 4l..4l+3) and write.
// ============================================================
#define MT  256          // vocab rows per block
#define KC  64           // K chunk staged in LDS
#define QST 512          // floats per K-quad row (BD*4)
#define LDS_FLOATS 8448  // epilogue 128*66 >= staging 16*512

__global__ __launch_bounds__(256) void k_gemm(const float* __restrict__ out_w,
                                              const float* __restrict__ out_b,
                                              const float* __restrict__ to_out,
                                              float* __restrict__ logits) {
    __shared__ float lds[LDS_FLOATS];
    const int tid = threadIdx.x;
    const int wave = tid >> 5;          // 0..7
    const int lane = tid & 31;
    const int lhalf = lane >> 4;        // selects K pair within quad
    const int l15 = lane & 15;
    const int vblock = blockIdx.x * MT;

    int r0 = vblock + wave * 16 + l15;
    if (r0 >= VD) r0 = VD - 1;          // clamp loads; stores guarded later
    int r1 = r0 + 128;
    if (r1 >= VD) r1 = VD - 1;
    const float* ap0 = out_w + (size_t)r0 * KOUT + 2 * lhalf;
    const float* ap1 = out_w + (size_t)r1 * KOUT + 2 * lhalf;

    v8f acc0[8] = {};
    v8f acc1[8] = {};

    for (int k0 = 0; k0 < KOUT; k0 += KC) {
        __syncthreads();
        // stage to_out[:, k0:k0+KC] in K-quad interleaved layout.
        // bit-sliced mapping keeps global reads coalesced per 4-k run and
        // LDS write banks = i (mod 64) -> conflict-free.
        for (int i = tid; i < KC * BD; i += 256) {
            int kc = (i & 3) | (((i >> 7) & 15) << 2);
            int bc = ((i >> 2) & 31) | (((i >> 11) & 3) << 5);
            lds[(kc >> 2) * QST + bc * 4 + (kc & 3)] =
                to_out[(size_t)bc * KOUT + k0 + kc];
        }
        __syncthreads();

        if (k0 + KC < KOUT) {           // pull next out_w chunk toward L0/L2
            __builtin_prefetch(ap0 + k0 + KC, 0, 3);
            __builtin_prefetch(ap1 + k0 + KC, 0, 3);
        }

        for (int kk = 0; kk < KC; kk += 4) {
            // A fragments: lane holds out_w[row, k0+kk+2*lhalf + {0,1}]
            v2f a0 = *(const v2f*)(ap0 + k0 + kk);
            v2f a1 = *(const v2f*)(ap1 + k0 + kk);
            // B fragments: rows kk+2*lhalf, +1 at col nt*16+l15, contiguous
            const float* bb = lds + (kk >> 2) * QST + l15 * 4 + 2 * lhalf;
            v2f bf[8];
            #pragma unroll
            for (int nt = 0; nt < 8; ++nt)
                bf[nt] = *(const v2f*)(bb + nt * 64);
            #pragma unroll
            for (int nt = 0; nt < 8; ++nt) {
                acc0[nt] = __builtin_amdgcn_wmma_f32_16x16x4_f32(
                    false, a0, false, bf[nt], (short)0, acc0[nt], false, false);
                acc1[nt] = __builtin_amdgcn_wmma_f32_16x16x4_f32(
                    false, a1, false, bf[nt], (short)0, acc1[nt], false, false);
            }
        }
    }

    // ---- epilogue: transpose through LDS, four 64-row phases ----
    #pragma unroll
    for (int p = 0; p < 4; ++p) {
        __syncthreads();
        if ((wave >> 2) == (p & 1)) {
            const v8f* accp = (p >= 2) ? acc1 : acc0;
            const int vbase = (wave & 3) * 16;
            #pragma unroll
            for (int nt = 0; nt < 8; ++nt) {
                const int bc = nt * 16 + l15;
                #pragma unroll
                for (int r = 0; r < 8; ++r) {
                    const int vl = vbase + r + 8 * lhalf;   // 0..63
                    lds[bc * 66 + vl] = accp[nt][r];
                }
            }
        }
        __syncthreads();
        for (int i = tid; i < BD * 64; i += 256) {
            const int vv = i & 63;
            const int bc = i >> 6;
            const int v = vblock + p * 64 + vv;
            if (v < VD)
                logits[(size_t)bc * VD + v] = lds[bc * 66 + vv] + out_b[v];
        }
    }
}

// ============================================================
extern "C" void kernel_launch(void* const* d_in, const int* in_sizes, int n_in,
                              void* d_out, int out_size, void* d_ws, size_t ws_size,
                              hipStream_t stream) {
    const int*   x      = (const int*)  d_in[0];
    const float* hidden = (const float*)d_in[1];
    const float* enc    = (const float*)d_in[2];
    const float* emb    = (const float*)d_in[3];
    const float* attn_w = (const float*)d_in[4];
    // d_in[5] = attn_b : provably cancels in softmax
    const float* vvec   = (const float*)d_in[6];
    const float* w_ih   = (const float*)d_in[7];
    const float* w_hh   = (const float*)d_in[8];
    const float* b_ih   = (const float*)d_in[9];
    const float* b_hh   = (const float*)d_in[10];
    const float* out_w  = (const float*)d_in[11];
    const float* out_b  = (const float*)d_in[12];

    float* ws     = (float*)d_ws;
    float* u2     = ws + WS_U2;
    float* ctx    = ws + WS_CTX;
    float* to_gru = ws + WS_TOGRU;
    float* to_out = ws + WS_TOOUT;

    float* logits = (float*)d_out;                    // [B, V]
    float* hnew   = (float*)d_out + (size_t)BD * VD;  // [1, B, H]

    k_u2  <<<K2H / 256, 256, 0, stream>>>(attn_w, vvec, u2);
    k_attn<<<BD, 256, 0, stream>>>(enc, u2, ctx);
    k_prep<<<BD, 256, 0, stream>>>(x, emb, ctx, to_gru, to_out);
    k_gru <<<BD, 512, 0, stream>>>(to_gru, hidden, w_ih, w_hh, b_ih, b_hh,
                                   to_out, hnew);
    k_gemm<<<(VD + MT - 1) / MT, 256, 0, stream>>>(out_w, out_b, to_out, logits);
}